// Linear_attention_10436770529651
// MI455X (gfx1250) — compile-verified
//
#include <hip/hip_runtime.h>
#include <math.h>

typedef float v2f __attribute__((ext_vector_type(2)));
typedef float v8f __attribute__((ext_vector_type(8)));

#define WMMA4(a, b, c) \
    __builtin_amdgcn_wmma_f32_16x16x4_f32(false, (a), false, (b), (short)0, (c), false, false)

__device__ __forceinline__ float leaky(float x) { return x >= 0.0f ? x : 0.1f * x; }
__device__ __forceinline__ float softplus(float x) { return log1pf(expf(x)); }

// ---------------------------------------------------------------------------
// Kernel 1: kp[b,n,c] = focus( leaky(k + pe2) / softplus(scale) )
// One block = 16 rows x 256 channels tile. grid = B*N/16 = 2048
// ---------------------------------------------------------------------------
__global__ __launch_bounds__(256) void prep_k_kernel(
    const float* __restrict__ k, const float* __restrict__ pe2,
    const float* __restrict__ scale, float* __restrict__ kp)
{
    __shared__ float klds[4096];
    __shared__ float fac[16];
    const int t   = threadIdx.x;
    const int blk = blockIdx.x;
    const int b   = blk >> 8;         // N/16 = 256 tiles per batch
    const int n0  = (blk & 255) * 16;

    const float* kbase = k   + ((size_t)(b * 4096 + n0)) * 256;
    const float* pbase = pe2 + (size_t)n0 * 256;

    // phase 1: load + pe + leaky + /softplus(scale) -> LDS (coalesced float4)
#pragma unroll
    for (int i = 0; i < 4; ++i) {
        int off = (i * 256 + t) * 4;                  // float offset in tile
        float4 kv4 = *(const float4*)(kbase + off);
        float4 p4  = *(const float4*)(pbase + off);
        int ch = off & 255;
        float4 s4 = *(const float4*)(scale + ch);
        float4 r;
        r.x = leaky(kv4.x + p4.x) / softplus(s4.x);
        r.y = leaky(kv4.y + p4.y) / softplus(s4.y);
        r.z = leaky(kv4.z + p4.z) / softplus(s4.z);
        r.w = leaky(kv4.w + p4.w) / softplus(s4.w);
        *(float4*)(klds + off) = r;
    }
    __syncthreads();

    // phase 2: per-row sum(x^2), sum(x^6); 16 threads per row, shfl reduce
    {
        int r  = t >> 4;
        int c0 = (t & 15) * 16;
        float s2 = 0.0f, s6 = 0.0f;
#pragma unroll
        for (int i = 0; i < 16; ++i) {
            float x  = klds[r * 256 + c0 + i];
            float x2 = x * x;
            s2 += x2;
            s6 += x2 * x2 * x2;
        }
#pragma unroll
        for (int m = 1; m < 16; m <<= 1) {
            s2 += __shfl_xor(s2, m, 32);
            s6 += __shfl_xor(s6, m, 32);
        }
        if ((t & 15) == 0) fac[r] = (s6 > 0.0f) ? sqrtf(s2 / s6) : 0.0f;
    }
    __syncthreads();

    // phase 3: kp = x^3 * fac[row], coalesced store
    float* obase = kp + ((size_t)(b * 4096 + n0)) * 256;
#pragma unroll
    for (int i = 0; i < 4; ++i) {
        int off = (i * 256 + t) * 4;
        int row = off >> 8;
        float f = fac[row];
        float4 x = *(const float4*)(klds + off);
        float4 o;
        o.x = x.x * x.x * x.x * f;
        o.y = x.y * x.y * x.y * f;
        o.z = x.z * x.z * x.z * f;
        o.w = x.w * x.w * x.w * f;
        *(float4*)(obase + off) = o;
    }
}

// ---------------------------------------------------------------------------
// Kernel 2a: partial kv. grid = (b,h) x 16 n-splits = 1024 blocks.
// Each block: 8 waves x 32 rows, wmma f32 16x16x4, cross-wave LDS reduction,
// partial 32x32 tile -> part[blk][e*32+d]
// ---------------------------------------------------------------------------
__global__ __launch_bounds__(256) void kv_partial_kernel(
    const float* __restrict__ kp, const float* __restrict__ v,
    float* __restrict__ part)
{
    __shared__ float red[8192];   // [wave][tile][lane][j]
    const int t     = threadIdx.x;
    const int w     = t >> 5;
    const int L     = t & 31;
    const int l16   = L & 15;
    const int kh    = L >> 4;     // K-half of the 16x4 / 4x16 operand layout
    const int blk   = blockIdx.x; // bh*16 + split
    const int bh    = blk >> 4;
    const int split = blk & 15;
    const int b     = bh >> 3, h = bh & 7;

    const float* kb = kp + ((size_t)(b * 4096 + split * 256)) * 256 + h * 32;
    const float* vb = v  + ((size_t)(b * 4096 + split * 256)) * 256 + h * 32;

    v8f acc0 = {}, acc1 = {}, acc2 = {}, acc3 = {};

    const int nBeg = w * 32, nEnd = nBeg + 32;
#pragma unroll
    for (int n0 = nBeg; n0 < nEnd; n0 += 4) {
        const float* kr = kb + (size_t)(n0 + 2 * kh) * 256;
        const float* vr = vb + (size_t)(n0 + 2 * kh) * 256;
        v2f a0, a1, b0, b1;
        // A (kp^T): M = d index (lanes), K = n index (vgpr + lane-half)
        a0.x = kr[l16];          a0.y = kr[256 + l16];
        a1.x = kr[16 + l16];     a1.y = kr[256 + 16 + l16];
        // B (v): N = e index (lanes), K = n index
        b0.x = vr[l16];          b0.y = vr[256 + l16];
        b1.x = vr[16 + l16];     b1.y = vr[256 + 16 + l16];
        acc0 = WMMA4(a0, b0, acc0);
        acc1 = WMMA4(a0, b1, acc1);
        acc2 = WMMA4(a1, b0, acc2);
        acc3 = WMMA4(a1, b1, acc3);
    }

#pragma unroll
    for (int j = 0; j < 8; ++j) {
        red[((w * 4 + 0) * 32 + L) * 8 + j] = acc0[j];
        red[((w * 4 + 1) * 32 + L) * 8 + j] = acc1[j];
        red[((w * 4 + 2) * 32 + L) * 8 + j] = acc2[j];
        red[((w * 4 + 3) * 32 + L) * 8 + j] = acc3[j];
    }
    __syncthreads();

    // reduce across 8 waves; write transposed (e-major) partial tile
    float* pb = part + (size_t)blk * 1024;
#pragma unroll
    for (int i = 0; i < 4; ++i) {
        int idx = i * 256 + t;          // ((tile*32 + lane)*8 + j)
        int j   = idx & 7;
        int L2  = (idx >> 3) & 31;
        int t2  = idx >> 8;             // tile = p*2 + q
        float s = 0.0f;
#pragma unroll
        for (int ww = 0; ww < 8; ++ww) s += red[((ww * 4 + t2) * 32 + L2) * 8 + j];
        int p = t2 >> 1, qq = t2 & 1;
        int dIdx = 16 * p + j + 8 * (L2 >> 4);    // C/D layout: M = j + 8*(lane/16)
        int e    = 16 * qq + (L2 & 15);
        pb[e * 32 + dIdx] = s;
    }
}

// ---------------------------------------------------------------------------
// Kernel 2b: kvT[bh][e][d] = (1/N) * sum over 16 splits. 65536 outputs.
// ---------------------------------------------------------------------------
__global__ __launch_bounds__(256) void kv_reduce_kernel(
    const float* __restrict__ part, float* __restrict__ kvT)
{
    int gid = blockIdx.x * 256 + threadIdx.x;   // 0..65535
    int bh  = gid >> 10;
    int idx = gid & 1023;
    float s = 0.0f;
#pragma unroll
    for (int sp = 0; sp < 16; ++sp)
        s += part[(((size_t)bh * 16 + sp) << 10) + idx];
    kvT[gid] = s * (1.0f / 4096.0f);
}

// ---------------------------------------------------------------------------
// Kernel 3: per (b, 16-row n tile):
//   qp = focus(leaky(q+pe1)/s)  (in LDS)
//   per head (one wave each): x = qp_h @ kv_h, cv = leaky(v_h @ w^T + bias)
//   out[b, 32h+o, n] = x + cv   (transposed through LDS, coalesced stores)
// ---------------------------------------------------------------------------
__global__ __launch_bounds__(256) void attn_out_kernel(
    const float* __restrict__ q, const float* __restrict__ v,
    const float* __restrict__ pe1, const float* __restrict__ scale,
    const float* __restrict__ w_conv, const float* __restrict__ b_conv,
    const float* __restrict__ kvT, float* __restrict__ out)
{
    __shared__ float qlds[4096];
    __shared__ float vlds[4096];
    __shared__ float fac[16];
    __shared__ float olds[4096];
    const int t   = threadIdx.x;
    const int blk = blockIdx.x;
    const int b   = blk >> 8;
    const int n0  = (blk & 255) * 16;

    const float* qbase = q   + ((size_t)(b * 4096 + n0)) * 256;
    const float* vbase = v   + ((size_t)(b * 4096 + n0)) * 256;
    const float* pbase = pe1 + (size_t)n0 * 256;

    // phase 1: stage transformed q and raw v
#pragma unroll
    for (int i = 0; i < 4; ++i) {
        int off = (i * 256 + t) * 4;
        float4 q4 = *(const float4*)(qbase + off);
        float4 p4 = *(const float4*)(pbase + off);
        int ch = off & 255;
        float4 s4 = *(const float4*)(scale + ch);
        float4 r;
        r.x = leaky(q4.x + p4.x) / softplus(s4.x);
        r.y = leaky(q4.y + p4.y) / softplus(s4.y);
        r.z = leaky(q4.z + p4.z) / softplus(s4.z);
        r.w = leaky(q4.w + p4.w) / softplus(s4.w);
        *(float4*)(qlds + off) = r;
        *(float4*)(vlds + off) = *(const float4*)(vbase + off);
    }
    __syncthreads();

    // phase 2: row norms of q
    {
        int r  = t >> 4;
        int c0 = (t & 15) * 16;
        float s2 = 0.0f, s6 = 0.0f;
#pragma unroll
        for (int i = 0; i < 16; ++i) {
            float x  = qlds[r * 256 + c0 + i];
            float x2 = x * x;
            s2 += x2;
            s6 += x2 * x2 * x2;
        }
#pragma unroll
        for (int m = 1; m < 16; m <<= 1) {
            s2 += __shfl_xor(s2, m, 32);
            s6 += __shfl_xor(s6, m, 32);
        }
        if ((t & 15) == 0) fac[r] = (s6 > 0.0f) ? sqrtf(s2 / s6) : 0.0f;
    }
    __syncthreads();

    // phase 3: qp = q^3 * fac, in place
#pragma unroll
    for (int i = 0; i < 4; ++i) {
        int off = (i * 256 + t) * 4;
        float f = fac[off >> 8];
        float4 x = *(const float4*)(qlds + off);
        float4 o;
        o.x = x.x * x.x * x.x * f;
        o.y = x.y * x.y * x.y * f;
        o.z = x.z * x.z * x.z * f;
        o.w = x.w * x.w * x.w * f;
        *(float4*)(qlds + off) = o;
    }
    __syncthreads();

    // per-wave head: wave w handles head h = w
    const int w = t >> 5, L = t & 31, l16 = L & 15, kh = L >> 4;
    const int h = w;
    const float* kvh = kvT + (size_t)(b * 8 + h) * 32 * 32;

    v8f cx0 = {}, cx1 = {}, cc0 = {}, cc1 = {};
#pragma unroll
    for (int k0 = 0; k0 < 32; k0 += 4) {
        int kk = k0 + 2 * kh;
        v2f aq  = *(const v2f*)(qlds + l16 * 256 + 32 * h + kk);   // A: qp tile
        v2f av  = *(const v2f*)(vlds + l16 * 256 + 32 * h + kk);   // A: v tile
        v2f bk0 = *(const v2f*)(kvh + (size_t)l16 * 32 + kk);      // B: kvT[e][d]
        v2f bk1 = *(const v2f*)(kvh + (size_t)(16 + l16) * 32 + kk);
        v2f bw0 = *(const v2f*)(w_conv + (size_t)l16 * 32 + kk);   // B: w[o][d]
        v2f bw1 = *(const v2f*)(w_conv + (size_t)(16 + l16) * 32 + kk);
        cx0 = WMMA4(aq, bk0, cx0);
        cx1 = WMMA4(aq, bk1, cx1);
        cc0 = WMMA4(av, bw0, cc0);
        cc1 = WMMA4(av, bw1, cc1);
    }

    // combine: out = x + leaky(cv + bias); stage transposed [o][m] in LDS
    float bc0 = b_conv[l16];
    float bc1 = b_conv[16 + l16];
    float* ob = olds + w * 512;
#pragma unroll
    for (int j = 0; j < 8; ++j) {
        int m = j + 8 * kh;
        ob[l16 * 16 + m]        = cx0[j] + leaky(cc0[j] + bc0);
        ob[(16 + l16) * 16 + m] = cx1[j] + leaky(cc1[j] + bc1);
    }
    __syncthreads();

    // coalesced stores: per iteration 8 channels x 64B, fully covered
    float* outb = out + ((size_t)(b * 256 + 32 * w)) * 4096 + n0;
#pragma unroll
    for (int i = 0; i < 4; ++i) {
        int ch   = i * 8 + (L >> 2);
        int part = (L & 3) * 4;
        float4 val = *(const float4*)(ob + ch * 16 + part);
        *(float4*)(outb + (size_t)ch * 4096 + part) = val;
    }
}

// ---------------------------------------------------------------------------
extern "C" void kernel_launch(void* const* d_in, const int* in_sizes, int n_in,
                              void* d_out, int out_size, void* d_ws, size_t ws_size,
                              hipStream_t stream) {
    const float* q      = (const float*)d_in[0];
    const float* k      = (const float*)d_in[1];
    const float* v      = (const float*)d_in[2];
    const float* pe1    = (const float*)d_in[3];
    const float* pe2    = (const float*)d_in[4];
    const float* scale  = (const float*)d_in[5];
    const float* w_conv = (const float*)d_in[6];
    const float* b_conv = (const float*)d_in[7];
    float* out = (float*)d_out;

    float* kp   = (float*)d_ws;                       // 8*4096*256 floats = 32 MiB
    float* part = kp + (size_t)8 * 4096 * 256;        // 1024*1024 floats = 4 MiB
    float* kvT  = part + (size_t)1024 * 1024;         // 64*32*32 floats

    prep_k_kernel<<<2048, 256, 0, stream>>>(k, pe2, scale, kp);
    kv_partial_kernel<<<1024, 256, 0, stream>>>(kp, v, part);
    kv_reduce_kernel<<<256, 256, 0, stream>>>(part, kvT);
    attn_out_kernel<<<2048, 256, 0, stream>>>(q, v, pe1, scale, w_conv, b_conv, kvT, out);
}